// GATlayer_26414048870741
// MI455X (gfx1250) — compile-verified
//
#include <hip/hip_runtime.h>
#include <hip/hip_bf16.h>
#include <math.h>

#define D_FEAT 128
#define H_HEADS 8
#define C_CH 16
#define NEG_SLOPE 0.2f
#define BN_EPS 1e-5f

typedef float v2f __attribute__((ext_vector_type(2)));
typedef float v8f __attribute__((ext_vector_type(8)));

// float atomic max via monotone integer mapping (sign-bit split, -0.0 safe)
__device__ __forceinline__ void atomicMaxFloat(float* addr, float val) {
  if ((__float_as_uint(val) >> 31) == 0u) {
    atomicMax((int*)addr, __float_as_int(val));       // non-negative floats: int order == float order
  } else {
    atomicMin((unsigned int*)addr, __float_as_uint(val)); // negative floats: reversed unsigned order
  }
}

// ---------------------------------------------------------------------------
// Kernel 1: fea = x @ W + b via V_WMMA_F32_16X16X4_F32, one 16x16 tile / wave.
// Col-tile j == head j (C=16), so per-node attention scores
//   s_dst[n,h] = fea[n, h*16:h*16+16] . att[h, :16]
//   s_src[n,h] = fea[n, h*16:h*16+16] . att[h, 16:]
// are fused into the epilogue with 16-lane shfl_xor reductions.
// ---------------------------------------------------------------------------
__global__ __launch_bounds__(32)
void gemm_att_kernel(const float* __restrict__ x, const float* __restrict__ w,
                     const float* __restrict__ bias, const float* __restrict__ att,
                     float* __restrict__ fea, float* __restrict__ s_dst,
                     float* __restrict__ s_src)
{
  const int lane = threadIdx.x;      // 0..31, wave32
  const int half = lane >> 4;        // 0 or 1
  const int l16  = lane & 15;
  const int tileRow = blockIdx.x;    // N/16 tiles of rows
  const int head    = blockIdx.y;    // 8 heads == 8 col tiles

  const int arowBase = tileRow * 16;
  const int col = head * C_CH + l16;

  v8f acc = {};
  #pragma unroll 4
  for (int k0 = 0; k0 < D_FEAT; k0 += 4) {
    const int kb = k0 + half * 2;
    // A 16x4 f32: lanes 0-15 hold K={k0,k0+1}, lanes 16-31 hold K={k0+2,k0+3}
    v2f a = *(const v2f*)(x + (size_t)(arowBase + l16) * D_FEAT + kb);
    // B 4x16 f32: mirrored layout, N striped across lanes
    v2f b;
    b.x = w[(size_t)kb       * (H_HEADS * C_CH) + col];
    b.y = w[(size_t)(kb + 1) * (H_HEADS * C_CH) + col];
    acc = __builtin_amdgcn_wmma_f32_16x16x4_f32(false, a, false, b, (short)0, acc,
                                                false, false);
  }

  const float bv = bias[col];
  const float ai = att[head * 2 * C_CH + l16];         // att[0,h,:C]  (x_i / dst)
  const float aj = att[head * 2 * C_CH + C_CH + l16];  // att[0,h,C:]  (x_j / src)

  #pragma unroll
  for (int v = 0; v < 8; ++v) {
    const int row = arowBase + v + half * 8;           // C/D layout: VGPR v, halves
    float val = acc[v] + bv;
    fea[(size_t)row * D_FEAT + col] = val;
    float sd = val * ai;
    float ss = val * aj;
    #pragma unroll
    for (int m = 8; m >= 1; m >>= 1) {
      sd += __shfl_xor(sd, m, 16);
      ss += __shfl_xor(ss, m, 16);
    }
    if (l16 == 0) {
      s_dst[(size_t)row * H_HEADS + head] = sd;
      s_src[(size_t)row * H_HEADS + head] = ss;
    }
  }
}

// ---------------------------------------------------------------------------
__global__ void init_kernel(float* seg_max, float* seg_sum, float* aggr,
                            float* stats, int nh, int nfeat)
{
  int i = blockIdx.x * blockDim.x + threadIdx.x;
  if (i < nfeat) aggr[i] = 0.0f;
  if (i < nh) { seg_max[i] = -INFINITY; seg_sum[i] = 0.0f; }
  if (i < 2 * D_FEAT) stats[i] = 0.0f;
}

// alpha = leaky(quant(s_dst[dst] + s_src[src])); running segment max
__global__ void edge_alpha_kernel(const int* __restrict__ ei,
                                  const float* __restrict__ s_dst,
                                  const float* __restrict__ s_src,
                                  const float* __restrict__ g_ptr,
                                  float* __restrict__ alpha,
                                  float* __restrict__ seg_max, int E)
{
  int e = blockIdx.x * blockDim.x + threadIdx.x;
  if (e >= E) return;
  int s = ei[e];
  int d = ei[E + e];
  float g = g_ptr[0];
  float inv_g = 1.0f / g;
  const float4* pd = (const float4*)(s_dst + (size_t)d * H_HEADS);
  const float4* ps = (const float4*)(s_src + (size_t)s * H_HEADS);
  float4 d0 = pd[0], d1 = pd[1], s0 = ps[0], s1 = ps[1];
  float a[8] = { d0.x + s0.x, d0.y + s0.y, d0.z + s0.z, d0.w + s0.w,
                 d1.x + s1.x, d1.y + s1.y, d1.z + s1.z, d1.w + s1.w };
  #pragma unroll
  for (int h = 0; h < H_HEADS; ++h) {
    float q = rintf(a[h] * inv_g);                 // round half-to-even
    q = fminf(fmaxf(q, -8.0f), 7.0f);              // clip to [QN, QP]
    float v = q * g;                               // STE forward value
    v = (v >= 0.0f) ? v : NEG_SLOPE * v;           // leaky relu
    a[h] = v;
    atomicMaxFloat(&seg_max[(size_t)d * H_HEADS + h], v);
  }
  float4* pa = (float4*)(alpha + (size_t)e * H_HEADS);
  pa[0] = make_float4(a[0], a[1], a[2], a[3]);
  pa[1] = make_float4(a[4], a[5], a[6], a[7]);
}

// ex = exp(alpha - seg_max[dst]); running segment sum
__global__ void edge_exp_kernel(const int* __restrict__ ei, float* __restrict__ alpha,
                                const float* __restrict__ seg_max,
                                float* __restrict__ seg_sum, int E)
{
  int e = blockIdx.x * blockDim.x + threadIdx.x;
  if (e >= E) return;
  int d = ei[E + e];
  #pragma unroll
  for (int h = 0; h < H_HEADS; ++h) {
    float m  = seg_max[(size_t)d * H_HEADS + h];
    float ex = expf(alpha[(size_t)e * H_HEADS + h] - m);
    alpha[(size_t)e * H_HEADS + h] = ex;
    atomicAdd(&seg_sum[(size_t)d * H_HEADS + h], ex);
  }
}

// coef = ex / (seg_sum[dst] + 1e-16)
__global__ void edge_coef_kernel(const int* __restrict__ ei, float* __restrict__ alpha,
                                 const float* __restrict__ seg_sum, int E)
{
  int e = blockIdx.x * blockDim.x + threadIdx.x;
  if (e >= E) return;
  int d = ei[E + e];
  #pragma unroll
  for (int h = 0; h < H_HEADS; ++h) {
    float s = seg_sum[(size_t)d * H_HEADS + h] + 1e-16f;
    alpha[(size_t)e * H_HEADS + h] /= s;
  }
}

// aggr[dst,f] += fea[src,f] * coef[e, f/16]; thread per (edge, feature)
__global__ void scatter_kernel(const int* __restrict__ ei, const float* __restrict__ fea,
                               const float* __restrict__ coef, float* __restrict__ aggr,
                               int E)
{
  long idx = (long)blockIdx.x * blockDim.x + threadIdx.x;
  if (idx >= (long)E * D_FEAT) return;
  int f = (int)(idx & (D_FEAT - 1));
  int e = (int)(idx >> 7);
  int s = ei[e];
  int d = ei[E + e];
  float c = coef[(size_t)e * H_HEADS + (f >> 4)];
  atomicAdd(&aggr[(size_t)d * D_FEAT + f], fea[(size_t)s * D_FEAT + f] * c);
}

// column-wise partial sums of (aggr+bias) and its square
#define STAT_ROWS 250
__global__ __launch_bounds__(D_FEAT)
void stats_kernel(const float* __restrict__ aggr, const float* __restrict__ bias,
                  float* __restrict__ stats, int n)
{
  int f  = threadIdx.x;
  int r0 = blockIdx.x * STAT_ROWS;
  int r1 = min(r0 + STAT_ROWS, n);
  float b = bias[f], s = 0.0f, sq = 0.0f;
  for (int r = r0; r < r1; ++r) {
    float v = aggr[(size_t)r * D_FEAT + f] + b;
    s  += v;
    sq += v * v;
  }
  atomicAdd(&stats[f], s);
  atomicAdd(&stats[D_FEAT + f], sq);
}

// batchnorm (batch stats) + relu + residual; bit_sum passthrough
__global__ void finalize_kernel(const float* __restrict__ aggr, const float* __restrict__ bias,
                                const float* __restrict__ stats, const float* __restrict__ gamma,
                                const float* __restrict__ beta, const float* __restrict__ x,
                                const float* __restrict__ bit_sum, float* __restrict__ out,
                                int n)
{
  long idx = (long)blockIdx.x * blockDim.x + threadIdx.x;
  long total = (long)n * D_FEAT;
  if (idx == 0) out[total] = bit_sum[0];   // second tuple element
  if (idx >= total) return;
  int f = (int)(idx & (D_FEAT - 1));
  float inv_n = 1.0f / (float)n;
  float mu  = stats[f] * inv_n;
  float var = stats[D_FEAT + f] * inv_n - mu * mu;     // biased var (ddof=0)
  float v = aggr[idx] + bias[f];
  float y = (v - mu) * rsqrtf(var + BN_EPS) * gamma[f] + beta[f];
  y = fmaxf(y, 0.0f);
  out[idx] = y + x[idx];
}

// ---------------------------------------------------------------------------
extern "C" void kernel_launch(void* const* d_in, const int* in_sizes, int n_in,
                              void* d_out, int out_size, void* d_ws, size_t ws_size,
                              hipStream_t stream)
{
  const float* x          = (const float*)d_in[0];
  const int*   ei         = (const int*)d_in[1];
  const float* bit_sum    = (const float*)d_in[2];
  const float* lin_w      = (const float*)d_in[3];
  const float* lin_b      = (const float*)d_in[4];
  const float* att        = (const float*)d_in[5];
  const float* conv_bias  = (const float*)d_in[6];
  const float* bn_gamma   = (const float*)d_in[7];
  const float* bn_beta    = (const float*)d_in[8];
  const float* edge_gamma = (const float*)d_in[9];

  const int n = in_sizes[0] / D_FEAT;   // 50000
  const int e = in_sizes[1] / 2;        // 640000

  float* ws = (float*)d_ws;
  size_t o = 0;
  float* fea     = ws + o; o += (size_t)n * D_FEAT;
  float* s_dst   = ws + o; o += (size_t)n * H_HEADS;
  float* s_src   = ws + o; o += (size_t)n * H_HEADS;
  float* alpha   = ws + o; o += (size_t)e * H_HEADS;
  float* seg_max = ws + o; o += (size_t)n * H_HEADS;
  float* seg_sum = ws + o; o += (size_t)n * H_HEADS;
  float* aggr    = ws + o; o += (size_t)n * D_FEAT;
  float* stats   = ws + o; // 2*D_FEAT floats

  int total_nf = n * D_FEAT;
  init_kernel<<<(total_nf + 255) / 256, 256, 0, stream>>>(seg_max, seg_sum, aggr,
                                                          stats, n * H_HEADS, total_nf);

  dim3 g1(n / 16, H_HEADS);
  gemm_att_kernel<<<g1, 32, 0, stream>>>(x, lin_w, lin_b, att, fea, s_dst, s_src);

  int eb = (e + 255) / 256;
  edge_alpha_kernel<<<eb, 256, 0, stream>>>(ei, s_dst, s_src, edge_gamma, alpha, seg_max, e);
  edge_exp_kernel<<<eb, 256, 0, stream>>>(ei, alpha, seg_max, seg_sum, e);
  edge_coef_kernel<<<eb, 256, 0, stream>>>(ei, alpha, seg_sum, e);

  long total_ef = (long)e * D_FEAT;
  scatter_kernel<<<(int)((total_ef + 255) / 256), 256, 0, stream>>>(ei, fea, alpha, aggr, e);

  stats_kernel<<<(n + STAT_ROWS - 1) / STAT_ROWS, D_FEAT, 0, stream>>>(aggr, conv_bias, stats, n);

  finalize_kernel<<<(int)((total_nf + 255) / 256), 256, 0, stream>>>(
      aggr, conv_bias, stats, bn_gamma, bn_beta, x, bit_sum, (float*)d_out, n);
}